// BaseTaskAlignedAssigner_80788334838110
// MI455X (gfx1250) — compile-verified
//
#include <hip/hip_runtime.h>
#include <hip/hip_bf16.h>
#include <stdint.h>

// TaskAlignedAssigner for MI455X (gfx1250, wave32).
// B=8, L=33600, N=128, C=80, TOPK=13, ALPHA=1, BETA=6.
#define BB 8
#define LL 33600
#define NN 128
#define CC 80
#define KK 13
#define EPSF 1e-9f

__device__ __forceinline__ float iou_xyxy(float gx0, float gy0, float gx1, float gy1,
                                          float area_g,
                                          float px0, float py0, float px1, float py1,
                                          float area_p) {
  float ltx = fmaxf(gx0, px0), lty = fmaxf(gy0, py0);
  float rbx = fminf(gx1, px1), rby = fminf(gy1, py1);
  float w = fmaxf(rbx - ltx, 0.0f), h = fmaxf(rby - lty, 0.0f);
  float inter = w * h;
  float iou = inter / (area_g + area_p - inter + EPSF);
  iou = (iou > 1.0f + EPSF) ? 0.0f : iou;       // reference "where(iou>1+eps, 0, iou)"
  iou = fminf(fmaxf(iou, 0.0f), 1.0f);          // nan_to_num + clip (fmaxf(NaN,0)=0)
  return iou;
}

// ---------------------------------------------------------------------------
// Stage 1: per-(b,n) top-13 over L anchors of s = class_score * iou^6 * spatial.
// Stable ties (value desc, index asc) to match jax.lax.top_k exactly.
// Scatters bit n into per-anchor 128-bit positive mask (spatial & pad applied).
// ---------------------------------------------------------------------------
__global__ void taa_topk_kernel(const float* __restrict__ pred_scores,
                                const float* __restrict__ pred_boxes,
                                const float* __restrict__ anchors,
                                const int* __restrict__ gt_labels,
                                const float* __restrict__ gt_boxes,
                                const float* __restrict__ pad_gt_mask,
                                unsigned* __restrict__ mask) {
  __shared__ float rv[256];
  __shared__ int ri[256];

  const int bn = blockIdx.x;            // [0, B*N)
  const int b = bn / NN;
  const int n = bn % NN;
  if (pad_gt_mask[bn] <= 0.0f) return;  // padded GT never becomes positive

  const float gx0 = gt_boxes[bn * 4 + 0], gy0 = gt_boxes[bn * 4 + 1];
  const float gx1 = gt_boxes[bn * 4 + 2], gy1 = gt_boxes[bn * 4 + 3];
  const float area_g = (gx1 - gx0) * (gy1 - gy0);
  int lbl = gt_labels[bn];
  lbl = lbl < 0 ? 0 : (lbl > CC - 1 ? CC - 1 : lbl);

  const float* ps = pred_scores + (size_t)b * LL * CC + lbl;      // gather column
  const float4* pb4 = (const float4*)pred_boxes + (size_t)b * LL; // [L] xyxy
  const float2* ap2 = (const float2*)anchors;                     // [L] xy

  float tv[KK];
  int ti[KK];
#pragma unroll
  for (int j = 0; j < KK; ++j) { tv[j] = -1.0f; ti[j] = 0x7fffffff; }

  for (int l = threadIdx.x; l < LL; l += 256) {
    __builtin_prefetch(&pb4[l + 1024], 0, 0);  // gfx1250 global_prefetch, speculative
    float4 p = pb4[l];
    float2 a = ap2[l];
    float s = 0.0f;
    if (a.x >= gx0 && a.x <= gx1 && a.y >= gy0 && a.y <= gy1) {
      float area_p = (p.z - p.x) * (p.w - p.y);
      float iou = iou_xyxy(gx0, gy0, gx1, gy1, area_g, p.x, p.y, p.z, p.w, area_p);
      float i2 = iou * iou;
      s = ps[(size_t)l * CC] * (i2 * i2 * i2);  // ALPHA=1, BETA=6
    }
    if (s > tv[KK - 1]) {  // strict: equal keeps earlier (smaller) index
      float cv = s;
      int ci = l;
#pragma unroll
      for (int j = 0; j < KK; ++j) {  // stable insertion, static indexing only
        bool take = (cv > tv[j]);
        float nv = take ? cv : tv[j];
        int ni = take ? ci : ti[j];
        float ov = take ? tv[j] : cv;
        int oi = take ? ti[j] : ci;
        tv[j] = nv; ti[j] = ni; cv = ov; ci = oi;
      }
    }
  }

  // 13 rounds of cross-thread argmax merge over each thread's sorted list.
  int cursor = 0;
  unsigned* mrow = mask + (size_t)b * LL * 4 + (n >> 5);
  const unsigned nbit = 1u << (n & 31);
  for (int r = 0; r < KK; ++r) {
    float pv = (cursor < KK) ? tv[cursor] : -2.0f;
    int pi = (cursor < KK) ? ti[cursor] : 0x7fffffff;
    rv[threadIdx.x] = pv;
    ri[threadIdx.x] = pi;
    __syncthreads();
    for (int st = 128; st > 0; st >>= 1) {
      if ((int)threadIdx.x < st) {
        float ov = rv[threadIdx.x + st];
        int oi = ri[threadIdx.x + st];
        float mv = rv[threadIdx.x];
        int mi = ri[threadIdx.x];
        if (ov > mv || (ov == mv && oi < mi)) { rv[threadIdx.x] = ov; ri[threadIdx.x] = oi; }
      }
      __syncthreads();
    }
    int wi = ri[0];
    __syncthreads();
    if (pi == wi && cursor < KK) cursor++;
    if (threadIdx.x == 0) {
      float2 a = ap2[wi];  // positive = topk & spatial & pad
      if (a.x >= gx0 && a.x <= gx1 && a.y >= gy0 && a.y <= gy1)
        atomicOr(&mrow[(size_t)wi * 4], nbit);
    }
  }
}

// ---------------------------------------------------------------------------
// Stage 2: per-anchor conflict resolution + assignment + per-GT atomic maxes.
// GT table (boxes+labels) staged into LDS via gfx1250 async global->LDS loads.
// ---------------------------------------------------------------------------
__global__ void taa_assign_kernel(const float* __restrict__ pred_scores,
                                  const float* __restrict__ pred_boxes,
                                  const float* __restrict__ anchors,
                                  const int* __restrict__ gt_labels,
                                  const float* __restrict__ gt_boxes,
                                  const int* __restrict__ bg_ptr,
                                  const unsigned* __restrict__ mask,
                                  float* __restrict__ out_labels,
                                  float* __restrict__ out_boxes,
                                  int* __restrict__ rec_idx,
                                  float* __restrict__ rec_align,
                                  int* __restrict__ max_align_bits,
                                  int* __restrict__ max_iou_bits) {
  __shared__ __align__(16) float sGT[NN * 4];
  __shared__ int sLab[NN];

  const int blocksPerBatch = (LL + 255) / 256;
  const int b = blockIdx.x / blocksPerBatch;
  const int l = (blockIdx.x % blocksPerBatch) * 256 + threadIdx.x;
  const int tid = threadIdx.x;

  const float* gb = gt_boxes + (size_t)b * NN * 4;
  const int* gl = gt_labels + (size_t)b * NN;

  // Async copy GT boxes (128 x b128) and labels (128 x b32) into LDS.
  if (tid < NN) {
    unsigned ldsBox = (unsigned)(uintptr_t)(&sGT[tid * 4]);
    unsigned ldsLab = (unsigned)(uintptr_t)(&sLab[tid]);
    unsigned voffBox = (unsigned)tid * 16u;
    unsigned voffLab = (unsigned)tid * 4u;
    asm volatile("global_load_async_to_lds_b128 %0, %1, %2"
                 :
                 : "v"(ldsBox), "v"(voffBox), "s"(gb)
                 : "memory");
    asm volatile("global_load_async_to_lds_b32 %0, %1, %2"
                 :
                 : "v"(ldsLab), "v"(voffLab), "s"(gl)
                 : "memory");
  }
  asm volatile("s_wait_asynccnt 0" ::: "memory");
  __syncthreads();

  if (l >= LL) return;
  const size_t al = (size_t)b * LL + l;

  const float4 p = ((const float4*)pred_boxes)[al];
  const float2 a = ((const float2*)anchors)[l];
  const float area_p = (p.z - p.x) * (p.w - p.y);
  (void)a;

  const unsigned* mw = mask + al * 4;
  unsigned m0 = mw[0], m1 = mw[1], m2 = mw[2], m3 = mw[3];
  const int cnt = __popc(m0) + __popc(m1) + __popc(m2) + __popc(m3);

  // best_gt = argmax_n iou (first occurrence), over ALL 128 GTs (incl. padded).
  float best = -1.0f;
  int bestn = 0;
  for (int g = 0; g < NN; ++g) {
    float4 gq = *(const float4*)&sGT[g * 4];
    float area_g = (gq.z - gq.x) * (gq.w - gq.y);
    float iou = iou_xyxy(gq.x, gq.y, gq.z, gq.w, area_g, p.x, p.y, p.z, p.w, area_p);
    if (iou > best) { best = iou; bestn = g; }
  }

  if (cnt > 1) {  // conflict: keep only the best-IoU GT's bit
    unsigned bb2 = 1u << (bestn & 31);
    int bw = bestn >> 5;
    m0 &= (bw == 0) ? bb2 : 0u;
    m1 &= (bw == 1) ? bb2 : 0u;
    m2 &= (bw == 2) ? bb2 : 0u;
    m3 &= (bw == 3) ? bb2 : 0u;
  }

  const bool has = (m0 | m1 | m2 | m3) != 0u;
  int gt = 0;  // argmax(positive) = lowest set bit; 0 when none (matches argmax of zeros)
  if (m0) gt = __ffs(m0) - 1;
  else if (m1) gt = 31 + __ffs(m1);
  else if (m2) gt = 63 + __ffs(m2);
  else if (m3) gt = 95 + __ffs(m3);

  const int lab = sLab[gt];
  out_labels[al] = has ? (float)lab : (float)(*bg_ptr);
  const float4 gsel = *(const float4*)&sGT[gt * 4];
  ((float4*)out_boxes)[al] = gsel;  // reference gathers box at gt_idx even if unassigned

  float a_val = 0.0f;
  if (has) {
    float area_g = (gsel.z - gsel.x) * (gsel.w - gsel.y);
    float iou_g = iou_xyxy(gsel.x, gsel.y, gsel.z, gsel.w, area_g, p.x, p.y, p.z, p.w, area_p);
    int lc = lab < 0 ? 0 : (lab > CC - 1 ? CC - 1 : lab);
    float cs = pred_scores[al * CC + lc];
    float i2 = iou_g * iou_g;
    a_val = cs * (i2 * i2 * i2);
    // non-negative floats: int-bitwise max == float max
    atomicMax(&max_align_bits[b * NN + gt], __float_as_int(a_val));
    atomicMax(&max_iou_bits[b * NN + gt], __float_as_int(iou_g));
  }
  rec_idx[al] = has ? gt : -1;
  rec_align[al] = a_val;
}

// ---------------------------------------------------------------------------
// Stage 3: normalized score scatter into pre-zeroed [B,L,C] output.
// ---------------------------------------------------------------------------
__global__ void taa_final_kernel(const int* __restrict__ rec_idx,
                                 const float* __restrict__ rec_align,
                                 const int* __restrict__ max_align_bits,
                                 const int* __restrict__ max_iou_bits,
                                 const int* __restrict__ gt_labels,
                                 float* __restrict__ out_scores) {
  size_t al = (size_t)blockIdx.x * 256 + threadIdx.x;
  if (al >= (size_t)BB * LL) return;
  int gt = rec_idx[al];
  if (gt < 0) return;  // bg: one_hot(C) row stays all-zero
  int b = (int)(al / LL);
  float ma = __int_as_float(max_align_bits[b * NN + gt]);
  float mi = __int_as_float(max_iou_bits[b * NN + gt]);
  float nf = rec_align[al] / (ma + EPSF) * mi;
  int lab = gt_labels[b * NN + gt];
  if (lab >= 0 && lab < CC) out_scores[al * CC + lab] = nf;
}

// ---------------------------------------------------------------------------
extern "C" void kernel_launch(void* const* d_in, const int* in_sizes, int n_in,
                              void* d_out, int out_size, void* d_ws, size_t ws_size,
                              hipStream_t stream) {
  (void)in_sizes; (void)n_in; (void)out_size; (void)ws_size;

  const float* pred_scores = (const float*)d_in[0];  // [B,L,C]
  const float* pred_boxes  = (const float*)d_in[1];  // [B,L,4]
  const float* anchors     = (const float*)d_in[2];  // [1,L,2]
  const int*   gt_labels   = (const int*)d_in[3];    // [B,N,1]
  const float* gt_boxes    = (const float*)d_in[4];  // [B,N,4]
  const float* pad_gt_mask = (const float*)d_in[5];  // [B,N,1]
  const int*   bg_ptr      = (const int*)d_in[6];    // scalar

  float* out = (float*)d_out;
  float* out_labels = out;                          // [B,L]
  float* out_boxes  = out + (size_t)BB * LL;        // [B,L,4]
  float* out_scores = out + (size_t)BB * LL * 5;    // [B,L,C]

  char* ws = (char*)d_ws;
  unsigned* mask = (unsigned*)ws;                                   // [B*L][4] bitmask
  size_t off = (size_t)BB * LL * 4 * sizeof(unsigned);
  int* rec_idx = (int*)(ws + off);      off += (size_t)BB * LL * sizeof(int);
  float* rec_align = (float*)(ws + off); off += (size_t)BB * LL * sizeof(float);
  int* max_a = (int*)(ws + off);        off += (size_t)BB * NN * sizeof(int);
  int* max_i = (int*)(ws + off);        off += (size_t)BB * NN * sizeof(int);

  hipMemsetAsync(mask, 0, (size_t)BB * LL * 4 * sizeof(unsigned), stream);
  hipMemsetAsync(max_a, 0, (size_t)BB * NN * sizeof(int), stream);
  hipMemsetAsync(max_i, 0, (size_t)BB * NN * sizeof(int), stream);
  hipMemsetAsync(out_scores, 0, (size_t)BB * LL * CC * sizeof(float), stream);

  taa_topk_kernel<<<BB * NN, 256, 0, stream>>>(pred_scores, pred_boxes, anchors,
                                               gt_labels, gt_boxes, pad_gt_mask, mask);

  const int bpb = (LL + 255) / 256;
  taa_assign_kernel<<<BB * bpb, 256, 0, stream>>>(pred_scores, pred_boxes, anchors,
                                                  gt_labels, gt_boxes, bg_ptr, mask,
                                                  out_labels, out_boxes,
                                                  rec_idx, rec_align, max_a, max_i);

  const size_t tot = (size_t)BB * LL;
  taa_final_kernel<<<(unsigned)((tot + 255) / 256), 256, 0, stream>>>(
      rec_idx, rec_align, max_a, max_i, gt_labels, out_scores);
}